// GraphCNN_18708877541515
// MI455X (gfx1250) — compile-verified
//
#include <hip/hip_runtime.h>

typedef __attribute__((ext_vector_type(2))) float v2f;
typedef __attribute__((ext_vector_type(8))) float v8f;

#define N_FEAT 128
#define XPAD 132            // padded LDS row stride for x tile (bank-conflict free A reads)
#define ROWS_PER_BLOCK 64   // 4 row-tiles of 16 per block
#define N_RT (ROWS_PER_BLOCK / 16)

// ---------------- kernel 1: zero output, deg[i] = 1 (self-loop) ----------------
__global__ void k_init(float* __restrict__ out, float* __restrict__ deg,
                       long long nOut, int nNodes) {
    long long i = (long long)blockIdx.x * blockDim.x + threadIdx.x;
    if (i < nOut) out[i] = 0.0f;
    if (i < nNodes) deg[i] = 1.0f;
}

// ---------------- kernel 2: degree accumulation over edges ----------------
__global__ void k_degree(const int* __restrict__ dst, float* __restrict__ deg, int nE) {
    int e = blockIdx.x * blockDim.x + threadIdx.x;
    if (e < nE) atomicAdd(&deg[dst[e]], 1.0f);
}

// ---------------- kernel 3: dinv = rsqrt(deg) in place ----------------
__global__ void k_rsqrt(float* __restrict__ deg, int n) {
    int i = blockIdx.x * blockDim.x + threadIdx.x;
    if (i < n) deg[i] = rsqrtf(deg[i]);   // deg >= 1 always (self-loop)
}

// ---------------- kernel 4: h = x @ W via V_WMMA_F32_16X16X4_F32 ----------------
// Block = 256 threads = 8 waves. Wave w owns column slab [16w, 16w+16).
// Block covers 64 rows (4 row-tiles of 16). W fully in LDS (64 KB), x tile padded.
// k is the OUTER loop: one B fragment per k-step is reused by all 4 row-tile
// accumulators -> 2.5 LDS loads per WMMA instead of 4.
__global__ void k_gemm_wmma(const float* __restrict__ x, const float* __restrict__ W,
                            float* __restrict__ h, int nNodes) {
    extern __shared__ float smem[];
    float* sW = smem;                      // 128*128 floats
    float* sX = smem + N_FEAT * N_FEAT;    // 64 * XPAD floats

    const int tid   = threadIdx.x;
    const int mBase = blockIdx.x * ROWS_PER_BLOCK;

    // stage W (row-major, 16384 floats) as float4
    {
        const float4* Wv  = (const float4*)W;
        float4*       sWv = (float4*)sW;
        for (int i = tid; i < (N_FEAT * N_FEAT) / 4; i += blockDim.x) sWv[i] = Wv[i];
    }
    // stage x tile (64 x 128) into padded LDS, zero-fill past end of matrix
    for (int i = tid; i < ROWS_PER_BLOCK * (N_FEAT / 4); i += blockDim.x) {
        int row = i >> 5;        // / (128/4)
        int c4  = i & 31;
        int g   = mBase + row;
        float4 v = {0.0f, 0.0f, 0.0f, 0.0f};
        if (g < nNodes) v = ((const float4*)(x + (size_t)g * N_FEAT))[c4];
        ((float4*)(sX + row * XPAD))[c4] = v;
    }
    __syncthreads();

    const int wave = tid >> 5;
    const int lane = tid & 31;
    const int half = lane >> 4;  // 0: K pair {0,1}, 1: K pair {2,3}
    const int lr   = lane & 15;
    const int n0   = wave * 16;

    v8f c0 = {}, c1 = {}, c2 = {}, c3 = {};
    const float* xbase = sX + lr * XPAD;   // A: row M = lr within each 16-row tile

    #pragma unroll
    for (int k0 = 0; k0 < N_FEAT; k0 += 4) {
        const int k = k0 + 2 * half;
        v2f b;
        b.x = sW[(size_t)k * N_FEAT + n0 + lr];        // B VGPR0: row K striped over lanes
        b.y = sW[(size_t)(k + 1) * N_FEAT + n0 + lr];  // B VGPR1: row K+1
        v2f a;
        a.x = xbase[k];            a.y = xbase[k + 1];
        c0 = __builtin_amdgcn_wmma_f32_16x16x4_f32(false, a, false, b, (short)0, c0, false, false);
        a.x = xbase[16 * XPAD + k];  a.y = xbase[16 * XPAD + k + 1];
        c1 = __builtin_amdgcn_wmma_f32_16x16x4_f32(false, a, false, b, (short)0, c1, false, false);
        a.x = xbase[32 * XPAD + k];  a.y = xbase[32 * XPAD + k + 1];
        c2 = __builtin_amdgcn_wmma_f32_16x16x4_f32(false, a, false, b, (short)0, c2, false, false);
        a.x = xbase[48 * XPAD + k];  a.y = xbase[48 * XPAD + k + 1];
        c3 = __builtin_amdgcn_wmma_f32_16x16x4_f32(false, a, false, b, (short)0, c3, false, false);
    }

    // C/D layout: VGPR r -> M = m0 + r (lanes 0-15) / m0 + 8 + r (lanes 16-31), N = n0+lr
    #pragma unroll
    for (int r = 0; r < 8; ++r) {
        int g0 = mBase + r + 8 * half;
        if (g0 < nNodes)      h[(size_t)g0 * N_FEAT + n0 + lr]        = c0[r];
        int g1 = g0 + 16;
        if (g1 < nNodes)      h[(size_t)g1 * N_FEAT + n0 + lr]        = c1[r];
        int g2 = g0 + 32;
        if (g2 < nNodes)      h[(size_t)g2 * N_FEAT + n0 + lr]        = c2[r];
        int g3 = g0 + 48;
        if (g3 < nNodes)      h[(size_t)g3 * N_FEAT + n0 + lr]        = c3[r];
    }
}

// ---------------- kernel 5: edge scatter. One wave per edge. ----------------
// Each lane handles 4 features: float4 gather of h[src], scale by norm,
// 4x global_atomic_add_f32 (no return) into out[dst].
__global__ void k_scatter(const float* __restrict__ h, const int* __restrict__ src,
                          const int* __restrict__ dst, const float* __restrict__ dinv,
                          float* __restrict__ out, int nE) {
    int wave = (int)(((long long)blockIdx.x * blockDim.x + threadIdx.x) >> 5);
    int lane = threadIdx.x & 31;
    if (wave >= nE) return;
    int s = src[wave];
    int d = dst[wave];
    float nrm = dinv[s] * dinv[d];
    float4 v = ((const float4*)(h + (size_t)s * N_FEAT))[lane];
    float* op = out + (size_t)d * N_FEAT + lane * 4;
    atomicAdd(op + 0, v.x * nrm);
    atomicAdd(op + 1, v.y * nrm);
    atomicAdd(op + 2, v.z * nrm);
    atomicAdd(op + 3, v.w * nrm);
}

// ---------------- kernel 6: self-loop + bias + ReLU ----------------
__global__ void k_finalize(float* __restrict__ out, const float* __restrict__ h,
                           const float* __restrict__ dinv, const float* __restrict__ b,
                           long long nOut) {
    long long i = (long long)blockIdx.x * blockDim.x + threadIdx.x;
    if (i >= nOut) return;
    int row = (int)(i >> 7);   // / 128
    int f   = (int)(i & 127);
    float di = dinv[row];
    float v = out[i] + h[i] * di * di + b[f];
    out[i] = fmaxf(v, 0.0f);
}

extern "C" void kernel_launch(void* const* d_in, const int* in_sizes, int n_in,
                              void* d_out, int out_size, void* d_ws, size_t ws_size,
                              hipStream_t stream) {
    const float* x  = (const float*)d_in[0];
    const float* W  = (const float*)d_in[1];
    const float* b  = (const float*)d_in[2];
    const int*   ei = (const int*)d_in[3];   // JAX x64 off -> int32; [2, E] flat
    float* out = (float*)d_out;

    const int nNodes = in_sizes[0] / N_FEAT;
    const int nE     = in_sizes[3] / 2;
    const int* src = ei;
    const int* dst = ei + nE;

    // workspace layout: deg/dinv [nNodes] (aligned), then h [nNodes * 128]
    float* deg = (float*)d_ws;
    float* h   = (float*)d_ws + (((size_t)nNodes + 1023) & ~(size_t)1023);

    const long long nOut = (long long)nNodes * N_FEAT;
    const int T = 256;

    k_init<<<(unsigned)((nOut + T - 1) / T), T, 0, stream>>>(out, deg, nOut, nNodes);
    k_degree<<<(nE + T - 1) / T, T, 0, stream>>>(dst, deg, nE);
    k_rsqrt<<<(nNodes + T - 1) / T, T, 0, stream>>>(deg, nNodes);

    const size_t smem = (N_FEAT * N_FEAT + ROWS_PER_BLOCK * XPAD) * sizeof(float); // ~97 KB < 320 KB WGP LDS
    k_gemm_wmma<<<(nNodes + ROWS_PER_BLOCK - 1) / ROWS_PER_BLOCK, T, smem, stream>>>(x, W, h, nNodes);

    // one wave (32 lanes) per edge -> 8 edges per 256-thread block
    k_scatter<<<(nE + 7) / 8, T, 0, stream>>>(h, src, dst, deg, out, nE);

    k_finalize<<<(unsigned)((nOut + T - 1) / T), T, 0, stream>>>(out, h, deg, b, nOut);
}